// MLPdirected_37632503447677
// MI455X (gfx1250) — compile-verified
//
#include <hip/hip_runtime.h>
#include <hip/hip_bf16.h>

#define BATCH   4
#define FMAPS   64
#define NNODE   512
#define NHID    32
#define W1COLS  129   // 2*FMAPS + 1

typedef __attribute__((ext_vector_type(2))) float v2f;
typedef __attribute__((ext_vector_type(8))) float v8f;

// ---------------------------------------------------------------------------
// Kernel A: sw[b,n] = (sum_i adj[b,i,n] - adj[b,n,n]) / (N-1)
// One thread per column; loads coalesced across n.
// ---------------------------------------------------------------------------
__global__ __launch_bounds__(256) void sw_kernel(const float* __restrict__ adj,
                                                 float* __restrict__ sw) {
    int tid = blockIdx.x * 256 + threadIdx.x;           // 0 .. 2047
    int b = tid >> 9;
    int n = tid & (NNODE - 1);
    const float* a = adj + (size_t)b * NNODE * NNODE;
    float s = 0.f;
    #pragma unroll 8
    for (int i = 0; i < NNODE; ++i)
        s += a[i * NNODE + n];
    s -= a[n * NNODE + n];
    sw[tid] = s * (1.0f / (float)(NNODE - 1));
}

// ---------------------------------------------------------------------------
// Kernel B: pa[b,n,h] = sum_f emb[b,n,f] * W1[h,f]            (h in 0..31)
//           pb[b,n,h] = sum_f emb[b,n,f] * W1[h,64+f] + sw[b,n]*W1[h,128] + b1[h]
// fp32 GEMM via V_WMMA_F32_16X16X4_F32. One wave -> 16 nodes x 32 h,
// 4 accumulators (pa/pb x two 16-wide h tiles), A fragment shared.
// emb_in layout: (B, F, N) so emb[b,n,f] = emb_in[b*F*N + f*N + n].
// ---------------------------------------------------------------------------
__global__ __launch_bounds__(256) void gemm_pab(const float* __restrict__ emb,
                                                const float* __restrict__ W1,
                                                const float* __restrict__ b1,
                                                const float* __restrict__ sw,
                                                float* __restrict__ pa,
                                                float* __restrict__ pb) {
    int lane = threadIdx.x & 31;
    int wave = threadIdx.x >> 5;
    int tile = blockIdx.x * 8 + wave;          // 128 tiles = (B*N)/16
    int b  = tile >> 5;
    int n0 = (tile & 31) << 4;

    const float* embB = emb + (size_t)b * FMAPS * NNODE;

    int row = n0 + (lane & 15);                // A: M = node
    int kh  = (lane >> 4) * 2;                 // A/B: k pair within the k-quad
    int hA  = lane & 15;                       // B: N = h (tile 0), +16 for tile 1

    const float* r0 = W1 + hA * W1COLS;        // W1 row for h = hA
    const float* r1 = W1 + (hA + 16) * W1COLS; // W1 row for h = hA + 16

    v8f c0 = {}; v8f c1 = {}; v8f c2 = {}; v8f c3 = {};

    #pragma unroll
    for (int k0 = 0; k0 < FMAPS; k0 += 4) {
        int k = k0 + kh;
        v2f a;
        a.x = embB[(k    ) * NNODE + row];
        a.y = embB[(k + 1) * NNODE + row];

        v2f ba0, ba1, bb0, bb1;
        ba0.x = r0[k];              ba0.y = r0[k + 1];            // W1a, h tile 0
        ba1.x = r1[k];              ba1.y = r1[k + 1];            // W1a, h tile 1
        bb0.x = r0[FMAPS + k];      bb0.y = r0[FMAPS + k + 1];    // W1b, h tile 0
        bb1.x = r1[FMAPS + k];      bb1.y = r1[FMAPS + k + 1];    // W1b, h tile 1

        c0 = __builtin_amdgcn_wmma_f32_16x16x4_f32(false, a, false, ba0, (short)0, c0, false, false);
        c1 = __builtin_amdgcn_wmma_f32_16x16x4_f32(false, a, false, ba1, (short)0, c1, false, false);
        c2 = __builtin_amdgcn_wmma_f32_16x16x4_f32(false, a, false, bb0, (short)0, c2, false, false);
        c3 = __builtin_amdgcn_wmma_f32_16x16x4_f32(false, a, false, bb1, (short)0, c3, false, false);
    }

    // pb bias terms, then store per the C/D layout:
    //   element r of lane l -> node n0 + r + 8*(lane>>4), h = hA (+16)
    float w1w0 = r0[2 * FMAPS];
    float w1w1 = r1[2 * FMAPS];
    float bia0 = b1[hA];
    float bia1 = b1[hA + 16];
    int mBase = n0 + ((lane >> 4) << 3);

    #pragma unroll
    for (int r = 0; r < 8; ++r) {
        int n = mBase + r;
        float s = sw[b * NNODE + n];
        float pb0 = c2[r] + s * w1w0 + bia0;
        float pb1 = c3[r] + s * w1w1 + bia1;
        float* paRow = pa + ((size_t)(b * NNODE + n)) * NHID;
        float* pbRow = pb + ((size_t)(b * NNODE + n)) * NHID;
        paRow[hA]      = c0[r];
        paRow[hA + 16] = c1[r];
        pbRow[hA]      = pb0;
        pbRow[hA + 16] = pb1;
    }
}

// ---------------------------------------------------------------------------
// Kernel C: out[b,i,j] = sigmoid(b2 + sum_h w2[h] * relu(pa[b,i,h]+pb[b,j,h]))
// 16x16 output tile per 256-thread block; pa/pb tiles staged in LDS
// (row stride 33 floats to avoid 8-way bank conflicts on the pb read).
// ---------------------------------------------------------------------------
__global__ __launch_bounds__(256) void pair_kernel(const float* __restrict__ pa,
                                                   const float* __restrict__ pb,
                                                   const float* __restrict__ W2,
                                                   const float* __restrict__ b2,
                                                   float* __restrict__ out) {
    __shared__ float sA[16][33];
    __shared__ float sB[16][33];
    __shared__ float sw2[NHID];

    int t  = threadIdx.x;
    int b  = blockIdx.z;
    int i0 = blockIdx.x << 4;
    int j0 = blockIdx.y << 4;

    // cooperative tile loads: 512 floats each, 2 per thread
    for (int idx = t; idx < 16 * NHID; idx += 256) {
        sA[idx >> 5][idx & 31] = pa[((size_t)(b * NNODE + i0)) * NHID + idx];
        sB[idx >> 5][idx & 31] = pb[((size_t)(b * NNODE + j0)) * NHID + idx];
    }
    if (t < NHID) sw2[t] = W2[t];
    __syncthreads();

    int i = t >> 4;
    int j = t & 15;
    const float* Ai = &sA[i][0];
    const float* Bj = &sB[j][0];

    float acc = b2[0];
    #pragma unroll
    for (int h = 0; h < NHID; ++h)
        acc += sw2[h] * fmaxf(Ai[h] + Bj[h], 0.0f);

    float y = 1.0f / (1.0f + __expf(-acc));
    out[(size_t)b * NNODE * NNODE + (size_t)(i0 + i) * NNODE + (j0 + j)] = y;
}

// ---------------------------------------------------------------------------
extern "C" void kernel_launch(void* const* d_in, const int* in_sizes, int n_in,
                              void* d_out, int out_size, void* d_ws, size_t ws_size,
                              hipStream_t stream) {
    const float* adj = (const float*)d_in[0];   // (B, N, N)
    const float* emb = (const float*)d_in[1];   // (B, F, N)
    const float* W1  = (const float*)d_in[2];   // (32, 129)
    const float* b1  = (const float*)d_in[3];   // (32,)
    const float* W2  = (const float*)d_in[4];   // (1, 32)
    const float* b2  = (const float*)d_in[5];   // (1,)
    // d_in[6] = layer (unused by the reference math)

    float* pa = (float*)d_ws;                         // B*N*H = 65536 floats
    float* pb = pa + BATCH * NNODE * NHID;            // 65536 floats
    float* sw = pb + BATCH * NNODE * NHID;            // B*N   =  2048 floats
    float* out = (float*)d_out;                       // (B, N, N)

    sw_kernel<<<(BATCH * NNODE) / 256, 256, 0, stream>>>(adj, sw);
    gemm_pab<<<(BATCH * NNODE / 16) / 8, 256, 0, stream>>>(emb, W1, b1, sw, pa, pb);
    pair_kernel<<<dim3(NNODE / 16, NNODE / 16, BATCH), 256, 0, stream>>>(pa, pb, W2, b2, out);
}